// FlashKAN_48533130444924
// MI455X (gfx1250) — compile-verified
//
#include <hip/hip_runtime.h>

// FlashKAN on gfx1250: dense-expanded B-spline coefficient rows (K=68)
// contracted with f32 WMMA 16x16x4. W panels moved by the Tensor Data Mover
// (one tensor_load_to_lds per block per channel, double-buffered, TENSORcnt).

typedef __attribute__((ext_vector_type(2))) float v2f;
typedef __attribute__((ext_vector_type(8))) float v8f;
typedef __attribute__((ext_vector_type(4))) unsigned int v4u;
typedef __attribute__((ext_vector_type(8))) int v8i;
typedef __attribute__((ext_vector_type(4))) int v4i;

constexpr int KORD   = 4;
constexpr int GRID   = 64;
constexpr int IN_DIM = 256;
constexpr int OUT_DIM = 256;
constexpr int BATCH  = 512;
constexpr int NROWS  = GRID + KORD;   // 68 weight rows
constexpr int BT = 32;                // batch tile per block
constexpr int OT = 32;                // output tile per block
constexpr int WS_PITCH = OT + 2;      // 34: TDM pads 2 dwords per 32-dword row
constexpr int KSTEPS = NROWS / 4;     // 17 WMMA K-steps of 4

// Issue one TDM descriptor: copy w rows 0..67, channel c, cols o0..o0+31 to LDS at byte offset lds_off
// (2D tile, row stride IN_DIM*OUT_DIM elements, pad 2 dwords per 32-dword row)
__device__ __forceinline__ void tdm_w_panel(const float* __restrict__ w,
                                            unsigned int lds_off, int c, int o0)
{
    const unsigned long long gaddr =
        (unsigned long long)(size_t)(w + ((size_t)0 * IN_DIM + c) * OUT_DIM + o0);

    v4u g0;
    g0.x = 1u;                                  // count=1 (valid), no gather
    g0.y = lds_off;                             // LDS byte address
    g0.z = (unsigned int)(gaddr & 0xFFFFFFFFull);
    g0.w = (unsigned int)(gaddr >> 32) | (2u << 30);   // global_addr[56:32] | type=2

    v8i g1;
    //        data_size=4B  pad_enable  pad_interval=32dw  pad_amount=2dw
    g1[0] = (int)((2u << 16) | (1u << 20) | (4u << 22) | (1u << 25));
    g1[1] = (int)((unsigned)OT << 16);          // tensor_dim0[15:0]=32
    g1[2] = (int)((unsigned)NROWS << 16);       // tensor_dim0[31:16]=0 | tensor_dim1[15:0]=68
    g1[3] = (int)((unsigned)OT << 16);          // tensor_dim1[31:16]=0 | tile_dim0=32
    g1[4] = NROWS;                              // tile_dim1=68 | tile_dim2=0
    g1[5] = IN_DIM * OUT_DIM;                   // tensor_dim0_stride[31:0]=65536
    g1[6] = 0;                                  // stride[47:32]=0 | dim1_stride[15:0]=0
    g1[7] = 0;

    v4i gz4 = {0, 0, 0, 0};                     // groups 2/3 unused (2D tile)
    v8i gz8 = {0, 0, 0, 0, 0, 0, 0, 0};         // extra group (6-arg toolchain form)
    __builtin_amdgcn_tensor_load_to_lds(g0, g1, gz4, gz4, gz8, 0);
}

__global__ __launch_bounds__(128) void flashkan_wmma_kernel(
    const float* __restrict__ x, const float* __restrict__ w,
    const float* __restrict__ t, float* __restrict__ out)
{
    __shared__ float xs[BT * IN_DIM];            // 32 KB   x slab
    __shared__ float As[BT * NROWS];             // 8.5 KB  dense coefficient rows
    __shared__ float Ws[2][NROWS * WS_PITCH];    // 18.1 KB double-buffered W panel [r][o]
    __shared__ float ts[71];                     // knot vector

    const int tid  = threadIdx.x;
    const int lane = tid & 31;
    const int wave = tid >> 5;
    const int sb   = wave >> 1;                  // batch sub-tile (0..1)
    const int so   = wave & 1;                   // out   sub-tile (0..1)
    const int b0   = blockIdx.y * BT;
    const int o0   = blockIdx.x * OT;

    const unsigned int ws_lds0 = (unsigned int)(size_t)&Ws[0][0];  // low 32 bits of the flat address are the LDS offset
    const unsigned int ws_lds1 = (unsigned int)(size_t)&Ws[1][0];

    // ---- stage x slab (coalesced float4) and knots; prime TDM pipeline ----
    {
        const float4* x4  = (const float4*)(x + (size_t)b0 * IN_DIM);
        float4*       xs4 = (float4*)xs;
        #pragma unroll
        for (int e = 0; e < (BT * IN_DIM) / 4 / 128; ++e)
            xs4[tid + e * 128] = x4[tid + e * 128];
        if (tid < 71) ts[tid] = t[tid];
    }
    if (wave == 0)
        tdm_w_panel(w, ws_lds0, 0, o0);          // prologue: channel 0 -> buf 0
    __syncthreads();

    v8f accA = {};                               // even K-steps
    v8f accB = {};                               // odd  K-steps (breaks WMMA RAW chain)
    const int M  = lane & 15;
    const int hi = lane >> 4;                    // 0: lanes 0-15 (K+0,1), 1: lanes 16-31 (K+2,3)

    for (int c = 0; c < IN_DIM; ++c) {
        const int buf = c & 1;

        // ---- build dense coefficient row for one batch each (threads 0..31) ----
        if (tid < BT) {
            float* row = &As[tid * NROWS];
            float4 z = {0.f, 0.f, 0.f, 0.f};
            #pragma unroll
            for (int q = 0; q < KSTEPS; ++q) ((float4*)row)[q] = z;

            const float xv = xs[tid * IN_DIM + c];
            int m = (int)floorf((xv + 1.0f) * (GRID * 0.5f));
            m = min(GRID - 1, max(0, m));
            const int ik = (KORD - 1) + m;       // i in [3,66]

            // Cox-de Boor order 4
            float Nv[KORD];
            Nv[0] = 1.0f;
            #pragma unroll
            for (int p = 1; p < KORD; ++p) {
                float saved = 0.0f;
                float nv[KORD];
                #pragma unroll
                for (int r = 0; r < p; ++r) {
                    const float rv = ts[ik + r + 1] - xv;
                    const float lv = xv - ts[ik - (p - 1 - r)];
                    const float tmp = Nv[r] / (rv + lv);
                    nv[r] = saved + rv * tmp;
                    saved = lv * tmp;
                }
                nv[p] = saved;
                #pragma unroll
                for (int r = 0; r <= p; ++r) Nv[r] = nv[r];
            }
            row[m + 0] = Nv[0];
            row[m + 1] = Nv[1];
            row[m + 2] = Nv[2];
            row[m + 3] = Nv[3];
            row[NROWS - 1] = xv / (1.0f + __expf(-xv));   // silu -> last weight row
        }

        // ---- wave 0: launch next panel's DMA, drain this panel's DMA ----
        if (wave == 0) {
            if (c + 1 < IN_DIM) {
                tdm_w_panel(w, (buf ? ws_lds0 : ws_lds1), c + 1, o0);
                __builtin_amdgcn_s_wait_tensorcnt(1);   // in-order: channel c resident
            } else {
                __builtin_amdgcn_s_wait_tensorcnt(0);
            }
        }
        __syncthreads();   // publish As + W panel to all waves

        // ---- 16x16 out sub-tile: 17 x V_WMMA_F32_16X16X4_F32, 2 interleaved chains ----
        const float* wp = &Ws[buf][0];
        const int bcol = so * 16 + M;
        #pragma unroll
        for (int s = 0; s < KSTEPS; ++s) {
            const int kk = s * 4 + hi * 2;
            v2f a = *(const v2f*)&As[(sb * 16 + M) * NROWS + kk];
            v2f b;
            b.x = wp[kk * WS_PITCH + bcol];          // B[kk][N]
            b.y = wp[(kk + 1) * WS_PITCH + bcol];    // B[kk+1][N]
            if (s & 1)
                accB = __builtin_amdgcn_wmma_f32_16x16x4_f32(
                    false, a, false, b, (short)0, accB, false, false);
            else
                accA = __builtin_amdgcn_wmma_f32_16x16x4_f32(
                    false, a, false, b, (short)0, accA, false, false);
        }
        __syncthreads();   // guard next iteration's As rewrite / buffer reuse
    }

    // ---- writeback: C/D layout = 8 VGPRs, lane%16 = N, row = v + 8*(lane/16) ----
    const v8f acc = accA + accB;
    #pragma unroll
    for (int v = 0; v < 8; ++v) {
        const int row = v + 8 * hi;
        out[(size_t)(b0 + sb * 16 + row) * OUT_DIM + (o0 + so * 16 + M)] = acc[v];
    }
}

extern "C" void kernel_launch(void* const* d_in, const int* in_sizes, int n_in,
                              void* d_out, int out_size, void* d_ws, size_t ws_size,
                              hipStream_t stream) {
    const float* x = (const float*)d_in[0];   // (512, 256) f32
    const float* w = (const float*)d_in[1];   // (68, 256, 256) f32
    const float* t = (const float*)d_in[2];   // (71,) f32 knots
    (void)d_ws; (void)ws_size; (void)in_sizes; (void)n_in; (void)out_size;
    float* out = (float*)d_out;               // (512, 256) f32

    dim3 grid(OUT_DIM / OT, BATCH / BT);      // (8, 16) = 128 blocks
    flashkan_wmma_kernel<<<grid, 128, 0, stream>>>(x, w, t, out);
}